// GraphConvAttentionModelWithBackbone_39393440039298
// MI455X (gfx1250) — compile-verified
//
#include <hip/hip_runtime.h>

// ---------------- problem constants (match reference) ----------------
#define NGR   25                     // nodes per graph
#define GNUM  4096                   // graphs
#define NNODES (GNUM * NGR)          // 102400
#define EGR   400                    // edges per graph
#define ETOT  (GNUM * EGR)           // 1638400

typedef __attribute__((ext_vector_type(2))) float v2f;
typedef __attribute__((ext_vector_type(8))) float v8f;

// Monotonic float<->uint key for atomicMax on signed floats in LDS.
__device__ __forceinline__ unsigned fkey(float f) {
  unsigned u = __float_as_uint(f);
  return (u & 0x80000000u) ? ~u : (u | 0x80000000u);
}
__device__ __forceinline__ float funkey(unsigned k) {
  return (k & 0x80000000u) ? __uint_as_float(k & 0x7fffffffu)
                           : __uint_as_float(~k);
}

// ======================================================================
// Kernel A: xlr1 = x @ [Wl1 | Wr1]   (102400x128 @ 128x(64+64) -> Nx128)
// One wave computes one 16x16 fp32 tile via V_WMMA_F32_16X16X4_F32.
// 8 waves/block cover all 128 output columns for a 16-row block.
// ======================================================================
__global__ __launch_bounds__(256) void k_gemm_l1(
    const float* __restrict__ x, const float* __restrict__ Wl,
    const float* __restrict__ Wr, float* __restrict__ out) {
  const int lane = threadIdx.x & 31;
  const int wave = threadIdx.x >> 5;            // 0..7 -> column tile
  const int rb   = blockIdx.x << 4;             // 16-row block
  const int m    = lane & 15;
  const int hi   = lane >> 4;                   // lane half selects K pair
  const int colbase = wave << 4;
  const float* W  = (colbase < 64) ? Wl : Wr;   // uniform per wave
  const int    cw = (colbase & 63) + m;         // column inside W (ld=64)
  const float* arow = x + (size_t)(rb + m) * 128;
  v8f acc = {};
#pragma unroll
  for (int k = 0; k < 128; k += 4) {
    const int ka = k + 2 * hi;
    v2f a, b;
    a.x = arow[ka];
    a.y = arow[ka + 1];
    b.x = W[ka * 64 + cw];
    b.y = W[(ka + 1) * 64 + cw];
    acc = __builtin_amdgcn_wmma_f32_16x16x4_f32(false, a, false, b,
                                                (short)0, acc, false, false);
  }
#pragma unroll
  for (int i = 0; i < 8; ++i)
    out[(size_t)(rb + i + 8 * hi) * 128 + colbase + m] = acc[i];
}

// ======================================================================
// Kernel B: GATv2 layer 1, one workgroup per graph, all-LDS edge phase.
// xlr: Nx128 (cols 0..63 = xl, 64..127 = xr). Output h1: Nx64 (lrelu 0.01)
// ======================================================================
__global__ __launch_bounds__(256) void k_gat1(
    const float* __restrict__ xlr, const float* __restrict__ eattr,
    const long long* __restrict__ eidx, const float* __restrict__ We,
    const float* __restrict__ att, const float* __restrict__ bias,
    float* __restrict__ h1) {
  __shared__ float    sXL[NGR * 64], sXR[NGR * 64], sAcc[NGR * 64];
  __shared__ float    sLog[EGR * 2];
  __shared__ unsigned sMax[NGR * 2];
  __shared__ float    sSum[NGR * 2];
  __shared__ int      sSrc[EGR], sDst[EGR];
  __shared__ float    sWe[3 * 64], sAtt[64], sB[64];

  const int g = blockIdx.x, nb = g * NGR, eb = g * EGR, tid = threadIdx.x;

  for (int i = tid; i < NGR * 64; i += 256) {
    const int nd = i >> 6, c = i & 63;
    sXL[i]  = xlr[(size_t)(nb + nd) * 128 + c];
    sXR[i]  = xlr[(size_t)(nb + nd) * 128 + 64 + c];
    sAcc[i] = 0.f;
  }
  for (int e = tid; e < EGR; e += 256) {
    sSrc[e] = (int)(eidx[eb + e] - nb);
    sDst[e] = (int)(eidx[(size_t)ETOT + eb + e] - nb);
  }
  if (tid < 192) sWe[tid] = We[tid];
  if (tid < 64) { sAtt[tid] = att[tid]; sB[tid] = bias[tid]; }
  if (tid < NGR * 2) { sMax[tid] = 0u; sSum[tid] = 0.f; }
  __syncthreads();

  // pass 1: per-(edge,head) logits + segment max
  for (int t = tid; t < EGR * 2; t += 256) {
    const int e = t >> 1, h = t & 1, s = sSrc[e], d = sDst[e];
    const float a0 = eattr[(size_t)(eb + e) * 3 + 0];
    const float a1 = eattr[(size_t)(eb + e) * 3 + 1];
    const float a2 = eattr[(size_t)(eb + e) * 3 + 2];
    float L = 0.f;
    const int cb = h << 5;
#pragma unroll 8
    for (int c2 = 0; c2 < 32; ++c2) {
      const int c = cb + c2;
      float v = sXL[s * 64 + c] + sXR[d * 64 + c] +
                a0 * sWe[c] + a1 * sWe[64 + c] + a2 * sWe[128 + c];
      v = v > 0.f ? v : 0.2f * v;               // GATv2 lrelu(0.2)
      L += sAtt[c] * v;
    }
    sLog[t] = L;
    atomicMax(&sMax[d * 2 + h], fkey(L));
  }
  __syncthreads();
  // pass 2: exp + segment sum
  for (int t = tid; t < EGR * 2; t += 256) {
    const int e = t >> 1, h = t & 1, d = sDst[e];
    const float z = __expf(sLog[t] - funkey(sMax[d * 2 + h]));
    sLog[t] = z;
    atomicAdd(&sSum[d * 2 + h], z);
  }
  __syncthreads();
  // pass 2.5: normalize -> alpha
  for (int t = tid; t < EGR * 2; t += 256) {
    const int e = t >> 1, h = t & 1, d = sDst[e];
    sLog[t] = sLog[t] / sSum[d * 2 + h];
  }
  __syncthreads();
  // pass 3: weighted scatter-add of xl[src]
  for (int t = tid; t < EGR * 64; t += 256) {
    const int e = t >> 6, c = t & 63, h = c >> 5;
    atomicAdd(&sAcc[sDst[e] * 64 + c], sLog[e * 2 + h] * sXL[sSrc[e] * 64 + c]);
  }
  __syncthreads();
  // epilogue: + bias, outer lrelu(0.01)
  for (int i = tid; i < NGR * 64; i += 256) {
    float v = sAcc[i] + sB[i & 63];
    v = v > 0.f ? v : 0.01f * v;
    h1[(size_t)nb * 64 + i] = v;
  }
}

// ======================================================================
// Kernel C: xlr2 = h1 @ [Wl2 | pad | Wr2 | pad]  (Nx64 @ 64x64 -> Nx64)
// cols 0..29 = xl2, 32..61 = xr2, others zero.
// ======================================================================
__device__ __forceinline__ float w2get(const float* Wl, const float* Wr,
                                       int k, int c) {
  if (c < 30) return Wl[k * 30 + c];
  if (c >= 32 && c < 62) return Wr[k * 30 + c - 32];
  return 0.f;
}

__global__ __launch_bounds__(128) void k_gemm_l2(
    const float* __restrict__ h1, const float* __restrict__ Wl,
    const float* __restrict__ Wr, float* __restrict__ out) {
  const int lane = threadIdx.x & 31;
  const int wave = threadIdx.x >> 5;            // 0..3
  const int rb = blockIdx.x << 4;
  const int m = lane & 15, hi = lane >> 4;
  const int c = (wave << 4) + m;
  const float* arow = h1 + (size_t)(rb + m) * 64;
  v8f acc = {};
#pragma unroll
  for (int k = 0; k < 64; k += 4) {
    const int ka = k + 2 * hi;
    v2f a, b;
    a.x = arow[ka];
    a.y = arow[ka + 1];
    b.x = w2get(Wl, Wr, ka, c);
    b.y = w2get(Wl, Wr, ka + 1, c);
    acc = __builtin_amdgcn_wmma_f32_16x16x4_f32(false, a, false, b,
                                                (short)0, acc, false, false);
  }
#pragma unroll
  for (int i = 0; i < 8; ++i)
    out[(size_t)(rb + i + 8 * hi) * 64 + c] = acc[i];
}

// ======================================================================
// Kernel D: GATv2 layer 2 (H=3,C=10) per graph; writes padded g (4096x768)
// ======================================================================
__global__ __launch_bounds__(256) void k_gat2(
    const float* __restrict__ xlr, const float* __restrict__ eattr,
    const long long* __restrict__ eidx, const float* __restrict__ We,
    const float* __restrict__ att, const float* __restrict__ bias,
    float* __restrict__ gout) {
  __shared__ float    sXL[NGR * 30], sXR[NGR * 30], sAcc[NGR * 30];
  __shared__ float    sLog[EGR * 3];
  __shared__ unsigned sMax[NGR * 3];
  __shared__ float    sSum[NGR * 3];
  __shared__ int      sSrc[EGR], sDst[EGR];
  __shared__ float    sWe[90], sAtt[30], sB[30];

  const int g = blockIdx.x, nb = g * NGR, eb = g * EGR, tid = threadIdx.x;

  for (int i = tid; i < NGR * 30; i += 256) {
    const int nd = i / 30, c = i - nd * 30;
    sXL[i]  = xlr[(size_t)(nb + nd) * 64 + c];
    sXR[i]  = xlr[(size_t)(nb + nd) * 64 + 32 + c];
    sAcc[i] = 0.f;
  }
  for (int e = tid; e < EGR; e += 256) {
    sSrc[e] = (int)(eidx[eb + e] - nb);
    sDst[e] = (int)(eidx[(size_t)ETOT + eb + e] - nb);
  }
  if (tid < 90) sWe[tid] = We[tid];
  if (tid < 30) { sAtt[tid] = att[tid]; sB[tid] = bias[tid]; }
  if (tid < NGR * 3) { sMax[tid] = 0u; sSum[tid] = 0.f; }
  __syncthreads();

  for (int t = tid; t < EGR * 3; t += 256) {
    const int e = t / 3, h = t - e * 3, s = sSrc[e], d = sDst[e];
    const float a0 = eattr[(size_t)(eb + e) * 3 + 0];
    const float a1 = eattr[(size_t)(eb + e) * 3 + 1];
    const float a2 = eattr[(size_t)(eb + e) * 3 + 2];
    float L = 0.f;
    const int cb = h * 10;
#pragma unroll
    for (int c2 = 0; c2 < 10; ++c2) {
      const int c = cb + c2;
      float v = sXL[s * 30 + c] + sXR[d * 30 + c] +
                a0 * sWe[c] + a1 * sWe[30 + c] + a2 * sWe[60 + c];
      v = v > 0.f ? v : 0.2f * v;
      L += sAtt[c] * v;
    }
    sLog[t] = L;
    atomicMax(&sMax[d * 3 + h], fkey(L));
  }
  __syncthreads();
  for (int t = tid; t < EGR * 3; t += 256) {
    const int e = t / 3, h = t - e * 3, d = sDst[e];
    const float z = __expf(sLog[t] - funkey(sMax[d * 3 + h]));
    sLog[t] = z;
    atomicAdd(&sSum[d * 3 + h], z);
  }
  __syncthreads();
  for (int t = tid; t < EGR * 3; t += 256) {
    const int e = t / 3, h = t - e * 3, d = sDst[e];
    sLog[t] = sLog[t] / sSum[d * 3 + h];
  }
  __syncthreads();
  for (int t = tid; t < EGR * 30; t += 256) {
    const int e = t / 30, c = t - e * 30, h = c / 10;
    atomicAdd(&sAcc[sDst[e] * 30 + c], sLog[e * 3 + h] * sXL[sSrc[e] * 30 + c]);
  }
  __syncthreads();
  for (int i = tid; i < 768; i += 256) {
    float v = 0.f;
    if (i < NGR * 30) {
      const int c = i % 30;
      v = sAcc[i] + sB[c];
      v = v > 0.f ? v : 0.01f * v;
    }
    gout[(size_t)g * 768 + i] = v;      // zero-padded cols 750..767
  }
}

// ======================================================================
// Kernel E: g1 = lrelu(g @ fc1_w + fc1_b)   (4096x768 @ 750x64, K padded)
// ======================================================================
__global__ __launch_bounds__(128) void k_gemm_fc1(
    const float* __restrict__ gmat, const float* __restrict__ W,
    const float* __restrict__ bias, float* __restrict__ out) {
  const int lane = threadIdx.x & 31;
  const int wave = threadIdx.x >> 5;            // 0..3
  const int rb = blockIdx.x << 4;
  const int m = lane & 15, hi = lane >> 4;
  const int c = (wave << 4) + m;
  const float* arow = gmat + (size_t)(rb + m) * 768;
  v8f acc = {};
  for (int k = 0; k < 768; k += 4) {
    const int ka = k + 2 * hi;
    v2f a, b;
    a.x = arow[ka];
    a.y = arow[ka + 1];
    b.x = (ka < 750) ? W[ka * 64 + c] : 0.f;
    b.y = (ka + 1 < 750) ? W[(ka + 1) * 64 + c] : 0.f;
    acc = __builtin_amdgcn_wmma_f32_16x16x4_f32(false, a, false, b,
                                                (short)0, acc, false, false);
  }
  const float bc = bias[c];
#pragma unroll
  for (int i = 0; i < 8; ++i) {
    float v = acc[i] + bc;
    v = v > 0.f ? v : 0.01f * v;
    out[(size_t)(rb + i + 8 * hi) * 64 + c] = v;
  }
}

// ======================================================================
// Kernel F: g2 = lrelu(g1 @ fc2_w + fc2_b)   (4096x64 @ 64x32)
// ======================================================================
__global__ __launch_bounds__(64) void k_gemm_fc2(
    const float* __restrict__ gin, const float* __restrict__ W,
    const float* __restrict__ bias, float* __restrict__ out) {
  const int lane = threadIdx.x & 31;
  const int wave = threadIdx.x >> 5;            // 0..1
  const int rb = blockIdx.x << 4;
  const int m = lane & 15, hi = lane >> 4;
  const int c = (wave << 4) + m;
  const float* arow = gin + (size_t)(rb + m) * 64;
  v8f acc = {};
#pragma unroll
  for (int k = 0; k < 64; k += 4) {
    const int ka = k + 2 * hi;
    v2f a, b;
    a.x = arow[ka];
    a.y = arow[ka + 1];
    b.x = W[ka * 32 + c];
    b.y = W[(ka + 1) * 32 + c];
    acc = __builtin_amdgcn_wmma_f32_16x16x4_f32(false, a, false, b,
                                                (short)0, acc, false, false);
  }
  const float bc = bias[c];
#pragma unroll
  for (int i = 0; i < 8; ++i) {
    float v = acc[i] + bc;
    v = v > 0.f ? v : 0.01f * v;
    out[(size_t)(rb + i + 8 * hi) * 32 + c] = v;
  }
}

// Kernel G: out[g] = g2[g,:] . fc3_w + fc3_b
__global__ __launch_bounds__(256) void k_fc3(
    const float* __restrict__ g2, const float* __restrict__ w,
    const float* __restrict__ b, float* __restrict__ out) {
  const int i = blockIdx.x * blockDim.x + threadIdx.x;
  if (i >= GNUM) return;
  const float* r = g2 + (size_t)i * 32;
  float s = b[0];
#pragma unroll
  for (int j = 0; j < 32; ++j) s += r[j] * w[j];
  out[i] = s;
}

// ======================================================================
extern "C" void kernel_launch(void* const* d_in, const int* in_sizes, int n_in,
                              void* d_out, int out_size, void* d_ws,
                              size_t ws_size, hipStream_t stream) {
  const float*     x     = (const float*)d_in[0];
  const float*     eattr = (const float*)d_in[1];
  const long long* eidx  = (const long long*)d_in[2];
  const float* Wl1 = (const float*)d_in[3];
  const float* Wr1 = (const float*)d_in[4];
  const float* We1 = (const float*)d_in[5];
  const float* at1 = (const float*)d_in[6];
  const float* b1  = (const float*)d_in[7];
  const float* Wl2 = (const float*)d_in[8];
  const float* Wr2 = (const float*)d_in[9];
  const float* We2 = (const float*)d_in[10];
  const float* at2 = (const float*)d_in[11];
  const float* b2  = (const float*)d_in[12];
  const float* f1w = (const float*)d_in[13];
  const float* f1b = (const float*)d_in[14];
  const float* f2w = (const float*)d_in[15];
  const float* f2b = (const float*)d_in[16];
  const float* f3w = (const float*)d_in[17];
  const float* f3b = (const float*)d_in[18];
  float* out = (float*)d_out;

  // Workspace layout with deterministic reuse (~79 MB peak):
  //  region A [0, N*128): xlr1; later reused as xlr2 [0,N*64) + gmat
  //  region B [N*128, N*128+N*64): h1; later reused as g1 + g2
  float* ws   = (float*)d_ws;
  float* xlr1 = ws;
  float* h1   = ws + (size_t)NNODES * 128;
  float* xlr2 = ws;                              // reuse A (xlr1 dead)
  float* gmat = ws + (size_t)NNODES * 64;        // A tail, disjoint from xlr2
  float* g1   = h1;                              // reuse B (h1 dead)
  float* g2   = h1 + (size_t)GNUM * 64;

  k_gemm_l1 <<<NNODES / 16, 256, 0, stream>>>(x, Wl1, Wr1, xlr1);
  k_gat1    <<<GNUM, 256, 0, stream>>>(xlr1, eattr, eidx, We1, at1, b1, h1);
  k_gemm_l2 <<<NNODES / 16, 128, 0, stream>>>(h1, Wl2, Wr2, xlr2);
  k_gat2    <<<GNUM, 256, 0, stream>>>(xlr2, eattr, eidx, We2, at2, b2, gmat);
  k_gemm_fc1<<<GNUM / 16, 128, 0, stream>>>(gmat, f1w, f1b, g1);
  k_gemm_fc2<<<GNUM / 16, 64, 0, stream>>>(g1, f2w, f2b, g2);
  k_fc3     <<<GNUM / 256, 256, 0, stream>>>(g2, f3w, f3b, out);
}